// NlEncoder_51848845197325
// MI455X (gfx1250) — compile-verified
//
#include <hip/hip_runtime.h>
#include <hip/hip_bf16.h>

#define GDIM 128

typedef float v2f __attribute__((ext_vector_type(2)));
typedef float v8f __attribute__((ext_vector_type(8)));

// ---------- monotonic float <-> uint encoding for atomicMax on floats ----------
__device__ __forceinline__ unsigned encf(float f) {
    unsigned u = __float_as_uint(f);
    return (u >> 31) ? ~u : (u | 0x80000000u);
}
__device__ __forceinline__ float decf(unsigned e) {
    return __uint_as_float((e >> 31) ? (e & 0x7FFFFFFFu) : ~e);
}
#define ENC_NEG_INF 0x007FFFFFu   // encf(-inf)

// ---------------- per-layer init: zero agg/denom, reset emax ----------------
__global__ void init_layer_kernel(float* __restrict__ agg, unsigned* __restrict__ emax,
                                  float* __restrict__ denom, int N, int ND) {
    int i = blockIdx.x * blockDim.x + threadIdx.x;
    if (i < ND) agg[i] = 0.0f;
    if (i < N) { emax[i] = ENC_NEG_INF; denom[i] = 0.0f; }
}

// ---------------- GEMM: H[N x 128] = X[N x 128] @ W[128 x 128] (f32 WMMA) -----
// A fragments from xs[row][k..k+1] (contiguous ds_load_b64).
// B fragments from K-transposed W tile wsT[col][k..k+1] (contiguous ds_load_b64).
__global__ __launch_bounds__(256) void gat_gemm_kernel(const float* __restrict__ X,
                                                       const float* __restrict__ W,
                                                       float* __restrict__ H, int N) {
    __shared__ float xs[32][132];    // 32 rows of X, padded (conflict-free A reads)
    __shared__ float wsT[128][68];   // 64-deep K-chunk of W, transposed, padded

    const int tid = threadIdx.x;
    const int rowBase = blockIdx.x * 32;

    // cooperative load of X tile (zero-fill OOB rows)
    for (int i = 0; i < 4; ++i) {
        int fi = tid + i * 256;           // float4 id, 0..1023
        int r  = fi >> 5;                 // 32 float4 per row
        int c4 = (fi & 31) * 4;
        float4 v = make_float4(0.f, 0.f, 0.f, 0.f);
        int grow = rowBase + r;
        if (grow < N) v = *(const float4*)(X + (size_t)grow * GDIM + c4);
        xs[r][c4 + 0] = v.x; xs[r][c4 + 1] = v.y; xs[r][c4 + 2] = v.z; xs[r][c4 + 3] = v.w;
    }

    const int lane = tid & 31, wid = tid >> 5;
    const int rowTile = wid & 1;          // 2 row tiles of 16
    const int colPair = wid >> 1;         // 4 col-pairs, each = two 16-wide tiles
    const int half = lane >> 4;           // lane half selects K sublane (A/B frag layout)
    const int l16  = lane & 15;
    const int aRow = rowTile * 16 + l16;
    const int c0 = colPair * 32 + l16;        // first column tile
    const int c1 = colPair * 32 + 16 + l16;   // second column tile

    v8f acc0 = {}; v8f acc1 = {};

    for (int kc = 0; kc < GDIM; kc += 64) {
        // cooperative load of W K-chunk, transposed into wsT[col][k]
        for (int i = 0; i < 8; ++i) {
            int fi = tid + i * 256;       // 0..2047
            int r  = fi >> 5;             // k within chunk (0..63)
            int c4 = (fi & 31) * 4;       // column
            float4 v = *(const float4*)(W + (size_t)(kc + r) * GDIM + c4);
            wsT[c4 + 0][r] = v.x; wsT[c4 + 1][r] = v.y;
            wsT[c4 + 2][r] = v.z; wsT[c4 + 3][r] = v.w;
        }
        __syncthreads();

        for (int kk = 0; kk < 64; kk += 4) {
            const int kb = kk + half * 2; // lanes 0-15 -> k,k+1 ; lanes 16-31 -> k+2,k+3
            v2f a  = *(const v2f*)&xs[aRow][kc + kb];
            v2f b0 = *(const v2f*)&wsT[c0][kb];
            acc0 = __builtin_amdgcn_wmma_f32_16x16x4_f32(false, a, false, b0,
                                                         (short)0, acc0, false, false);
            v2f b1 = *(const v2f*)&wsT[c1][kb];
            acc1 = __builtin_amdgcn_wmma_f32_16x16x4_f32(false, a, false, b1,
                                                         (short)0, acc1, false, false);
        }
        __syncthreads();
    }

    // store D tiles (C/D layout: VGPR i -> M=i (lanes 0-15) / M=8+i (lanes 16-31))
    const int gr0 = rowBase + rowTile * 16;
    for (int i = 0; i < 8; ++i) {
        int grow = gr0 + i + half * 8;
        if (grow < N) {
            H[(size_t)grow * GDIM + c0] = acc0[i];
            H[(size_t)grow * GDIM + c1] = acc1[i];
        }
    }
}

// -------- per-node attention logits: as = h . a_src, ad = h . a_dst (wave/row)
__global__ void alphas_kernel(const float* __restrict__ h, const float* __restrict__ a_src,
                              const float* __restrict__ a_dst, float* __restrict__ as_,
                              float* __restrict__ ad_, int N) {
    int gt = blockIdx.x * blockDim.x + threadIdx.x;
    int row = gt >> 5, lane = gt & 31;
    if (row >= N) return;
    float s1 = 0.f, s2 = 0.f;
    for (int c = lane; c < GDIM; c += 32) {
        float hv = h[(size_t)row * GDIM + c];
        s1 += hv * a_src[c];
        s2 += hv * a_dst[c];
    }
    for (int off = 16; off > 0; off >>= 1) {
        s1 += __shfl_xor(s1, off, 32);
        s2 += __shfl_xor(s2, off, 32);
    }
    if (lane == 0) { as_[row] = s1; ad_[row] = s2; }
}

__device__ __forceinline__ void edge_endpoints(int i, int E, const int* __restrict__ src,
                                               const int* __restrict__ dst, int& s, int& d) {
    if (i < E) { s = src[i]; d = dst[i]; } else { s = d = i - E; }
}

// ---------------- edge pass 1: segment max of leaky_relu logits ----------------
__global__ void edge_max_kernel(const int* __restrict__ src, const int* __restrict__ dst,
                                const float* __restrict__ as_, const float* __restrict__ ad_,
                                unsigned* __restrict__ emax, int E, int N) {
    int i = blockIdx.x * blockDim.x + threadIdx.x;
    if (i >= E + N) return;
    int s, d; edge_endpoints(i, E, src, dst, s, d);
    float e = as_[s] + ad_[d];
    e = e > 0.f ? e : 0.2f * e;
    atomicMax(&emax[d], encf(e));
}

// ---------------- edge pass 2: segment sum of exp(e - emax) ----------------
__global__ void edge_sum_kernel(const int* __restrict__ src, const int* __restrict__ dst,
                                const float* __restrict__ as_, const float* __restrict__ ad_,
                                const unsigned* __restrict__ emax, float* __restrict__ denom,
                                int E, int N) {
    int i = blockIdx.x * blockDim.x + threadIdx.x;
    if (i >= E + N) return;
    int s, d; edge_endpoints(i, E, src, dst, s, d);
    float e = as_[s] + ad_[d];
    e = e > 0.f ? e : 0.2f * e;
    unsafeAtomicAdd(&denom[d], expf(e - decf(emax[d])));
}

// ---------------- edge pass 3: agg[d] += alpha * h[s]   (one wave32 per edge) --
__global__ void edge_agg_kernel(const int* __restrict__ src, const int* __restrict__ dst,
                                const float* __restrict__ as_, const float* __restrict__ ad_,
                                const unsigned* __restrict__ emax, const float* __restrict__ denom,
                                const float* __restrict__ h, float* __restrict__ agg,
                                int E, int N) {
    int gt = blockIdx.x * blockDim.x + threadIdx.x;
    int i = gt >> 5, lane = gt & 31;
    if (i >= E + N) return;
    int s, d; edge_endpoints(i, E, src, dst, s, d);
    float e = as_[s] + ad_[d];
    e = e > 0.f ? e : 0.2f * e;
    float alpha = expf(e - decf(emax[d])) / (denom[d] + 1e-16f);

    const int c = lane * 4;
    float4 hv = *(const float4*)(h + (size_t)s * GDIM + c);
    float* ap = agg + (size_t)d * GDIM + c;
    unsafeAtomicAdd(ap + 0, hv.x * alpha);
    unsafeAtomicAdd(ap + 1, hv.y * alpha);
    unsafeAtomicAdd(ap + 2, hv.z * alpha);
    unsafeAtomicAdd(ap + 3, hv.w * alpha);
}

// ---------------- x = relu(agg + b) ----------------
__global__ void finalize_kernel(const float* __restrict__ agg, const float* __restrict__ b,
                                float* __restrict__ x, int ND) {
    int i = blockIdx.x * blockDim.x + threadIdx.x;
    if (i >= ND) return;
    float v = agg[i] + b[i & (GDIM - 1)];
    x[i] = v > 0.f ? v : 0.f;
}

// ---------------- final-stage scalars init ----------------
__global__ void init_final_kernel(unsigned* gmax, float* gsum, float* loss) {
    *gmax = ENC_NEG_INF;
    *gsum = 0.0f;
    *loss = 0.0f;
}

// ---------------- li = x . w_lin + b_lin, masked; global max (wave/row) --------
__global__ void li_kernel(const float* __restrict__ x, const float* __restrict__ w_lin,
                          const float* __restrict__ b_lin, const int* __restrict__ resmask,
                          float* __restrict__ li, unsigned* __restrict__ gmax, int N) {
    int gt = blockIdx.x * blockDim.x + threadIdx.x;
    int row = gt >> 5, lane = gt & 31;
    if (row >= N) return;
    float s = 0.f;
    for (int c = lane; c < GDIM; c += 32) s += x[(size_t)row * GDIM + c] * w_lin[c];
    for (int off = 16; off > 0; off >>= 1) s += __shfl_xor(s, off, 32);
    if (lane == 0) {
        s += b_lin[0];
        if (resmask[row] == 0) s = -1.0e9f;
        li[row] = s;
        atomicMax(gmax, encf(s));
    }
}

// ---------------- unnormalized softmax + global sum ----------------
__global__ void expsum_kernel(const float* __restrict__ li, const unsigned* __restrict__ gmax,
                              float* __restrict__ pun, float* __restrict__ gsum, int N) {
    int i = blockIdx.x * blockDim.x + threadIdx.x;
    if (i >= N) return;
    float p = expf(li[i] - decf(*gmax));
    pun[i] = p;
    unsafeAtomicAdd(gsum, p);
}

// ---------------- normalize, write softmax, accumulate loss ----------------
__global__ void loss_softmax_kernel(const float* __restrict__ pun, const float* __restrict__ gsum,
                                    const float* __restrict__ res, float* __restrict__ sm_out,
                                    float* __restrict__ loss, int N) {
    int i = blockIdx.x * blockDim.x + threadIdx.x;
    int lane = threadIdx.x & 31;
    float lp = 0.f;
    if (i < N) {
        float p = pun[i] / (*gsum);
        sm_out[i] = p;
        float pc = fminf(fmaxf(p, 1e-10f), 1.0f);
        lp = -logf(pc) * res[i];
    }
    for (int off = 16; off > 0; off >>= 1) lp += __shfl_xor(lp, off, 32);
    if (lane == 0) unsafeAtomicAdd(loss, lp);
}

// ---------------- tail: resmask -> float, s = x[:K] ----------------
__global__ void tail_kernel(const int* __restrict__ resmask, const float* __restrict__ x,
                            float* __restrict__ rm_out, float* __restrict__ s_out,
                            int N, int KD) {
    int i = blockIdx.x * blockDim.x + threadIdx.x;
    if (i < N) rm_out[i] = (float)resmask[i];
    if (i < KD) s_out[i] = x[i];
}

extern "C" void kernel_launch(void* const* d_in, const int* in_sizes, int n_in,
                              void* d_out, int out_size, void* d_ws, size_t ws_size,
                              hipStream_t stream) {
    const int N  = in_sizes[0] / GDIM;
    const int E  = in_sizes[1] / 2;
    const long ND = (long)N * GDIM;
    const int K  = (int)((out_size - 1 - 2L * N - ND) / GDIM);

    const float* x_in  = (const float*)d_in[0];
    const int*   src   = (const int*)d_in[1];
    const int*   dst   = src + E;
    const float* res   = (const float*)d_in[2];
    const int*   rmask = (const int*)d_in[3];
    const float* W1    = (const float*)d_in[5];
    const float* asr1  = (const float*)d_in[6];
    const float* ads1  = (const float*)d_in[7];
    const float* b1    = (const float*)d_in[8];
    const float* W2    = (const float*)d_in[9];
    const float* asr2  = (const float*)d_in[10];
    const float* ads2  = (const float*)d_in[11];
    const float* b2    = (const float*)d_in[12];
    const float* wlin  = (const float*)d_in[13];
    const float* blin  = (const float*)d_in[14];

    float* out    = (float*)d_out;
    float* lossp  = out;                 // [0]
    float* sm_out = out + 1;             // [1 .. N]
    float* x_out  = out + 1 + N;         // [.. + N*128]  (holds x1, later x2)
    float* rm_out = x_out + ND;          // [.. + N]
    float* s_out  = rm_out + N;          // [.. + K*128]

    float* wsf   = (float*)d_ws;
    float* h     = wsf;                  // N*128
    float* agg   = h + ND;               // N*128
    float* as_   = agg + ND;             // N
    float* ad_   = as_ + N;              // N
    float* denom = ad_ + N;              // N
    float* liA   = denom + N;            // N
    float* pun   = liA + N;              // N
    unsigned* emax = (unsigned*)(pun + N);  // N
    float* gsum  = (float*)(emax + N);   // 1
    unsigned* gmax = (unsigned*)(gsum + 1); // 1

    const int T = E + N;
    const int gND   = (int)((ND + 255) / 256);
    const int gRowW = (N * 32 + 255) / 256;   // wave-per-row kernels
    const int gEdge = (T + 255) / 256;
    const int gEdgW = (int)(((long)T * 32 + 255) / 256);

    auto layer = [&](const float* xin, const float* W, const float* a_s, const float* a_d,
                     const float* b, float* xdst) {
        init_layer_kernel<<<gND, 256, 0, stream>>>(agg, emax, denom, N, (int)ND);
        gat_gemm_kernel<<<(N + 31) / 32, 256, 0, stream>>>(xin, W, h, N);
        alphas_kernel<<<gRowW, 256, 0, stream>>>(h, a_s, a_d, as_, ad_, N);
        edge_max_kernel<<<gEdge, 256, 0, stream>>>(src, dst, as_, ad_, emax, E, N);
        edge_sum_kernel<<<gEdge, 256, 0, stream>>>(src, dst, as_, ad_, emax, denom, E, N);
        edge_agg_kernel<<<gEdgW, 256, 0, stream>>>(src, dst, as_, ad_, emax, denom, h, agg, E, N);
        finalize_kernel<<<gND, 256, 0, stream>>>(agg, b, xdst, (int)ND);
    };

    layer(x_in,  W1, asr1, ads1, b1, x_out);   // x1 staged in x output region
    layer(x_out, W2, asr2, ads2, b2, x_out);   // x2 overwrites (read-before-write via stream order)

    init_final_kernel<<<1, 1, 0, stream>>>(gmax, gsum, lossp);
    li_kernel<<<gRowW, 256, 0, stream>>>(x_out, wlin, blin, rmask, liA, gmax, N);
    expsum_kernel<<<(N + 255) / 256, 256, 0, stream>>>(liA, gmax, pun, gsum, N);
    loss_softmax_kernel<<<(N + 255) / 256, 256, 0, stream>>>(pun, gsum, res, sm_out, lossp, N);

    const int KD = K * GDIM;
    const int M  = (N > KD ? N : KD);
    tail_kernel<<<(M + 255) / 256, 256, 0, stream>>>(rmask, x_out, rm_out, s_out, N, KD);
}